// ParallelMinGRU_20349555048809
// MI455X (gfx1250) — compile-verified
//
#include <hip/hip_runtime.h>

// Problem constants (match reference)
#define BB   8
#define SS   4096
#define DXc  512
#define DHc  512
#define NC   64     // scan chunks per sequence
#define CLEN 64     // chunk length = SS/NC

// GEMM tiling
#define BM  128     // block rows
#define BN  64      // block cols (4 x 16 per wave)
#define KC  32      // K chunk staged in LDS
#define LDA 36      // A tile stride: even (8B frag reads) and mult of 4 (16B stages)

typedef __attribute__((ext_vector_type(2))) float v2f;
typedef __attribute__((ext_vector_type(8))) float v8f;
typedef int v4i_vs __attribute__((vector_size(16)));   // matches builtin param type

#if __has_builtin(__builtin_amdgcn_global_load_async_to_lds_b128)
#define ASYNC_G2L 1
#else
#define ASYNC_G2L 0
#endif

// 16-byte global -> LDS copy. Async (ASYNCcnt-tracked, no VGPR staging) when
// the toolchain exposes the gfx1250 builtin; plain load+store otherwise.
__device__ __forceinline__ void copy16_g2l(const float* __restrict__ gsrc,
                                           float* __restrict__ ldst) {
#if ASYNC_G2L
  __builtin_amdgcn_global_load_async_to_lds_b128(
      (__attribute__((address_space(1))) v4i_vs*)gsrc,
      (__attribute__((address_space(3))) v4i_vs*)ldst, 0, 0);
#else
  const float4 v = *(const float4*)gsrc;
  ldst[0] = v.x; ldst[1] = v.y; ldst[2] = v.z; ldst[3] = v.w;
#endif
}
__device__ __forceinline__ void stage_fence() {
#if ASYNC_G2L
  asm volatile("s_wait_asynccnt 0" ::: "memory");
#endif
}

// ---------- math helpers (fp32, numerically stable) ----------
__device__ __forceinline__ float softplus_f(float x) {
  return fmaxf(x, 0.0f) + log1pf(__expf(-fabsf(x)));
}
__device__ __forceinline__ float log_g_f(float x) {
  return (x >= 0.0f) ? __logf(x + 0.5f) : -softplus_f(-x);
}
__device__ __forceinline__ float logaddexp_f(float a, float b) {
  float mx = fmaxf(a, b);
  float mn = fminf(a, b);
  float r  = mx + log1pf(__expf(mn - mx));
  return (mn == -INFINITY) ? mx : r;   // handles both/-inf and NaN from inf-inf
}

// ---------- kernel 1: fused gate GEMMs + log-space transforms ----------
__global__ __launch_bounds__(256) void gates_gemm_kernel(
    const float* __restrict__ x,  const float* __restrict__ Wz,
    const float* __restrict__ bz, const float* __restrict__ Wh,
    const float* __restrict__ bh, float* __restrict__ coeff,
    float* __restrict__ val) {
  __shared__ float As[BM * LDA];       // x tile, [row][k]
  __shared__ float Wzs[KC * BN];       // Wz tile, [k][n] (memory layout)
  __shared__ float Whs[KC * BN];       // Wh tile, [k][n]
  const int tid  = threadIdx.x;
  const int wave = tid >> 5;
  const int lane = tid & 31;
  const int lh   = lane & 15;          // N col / A row within 16-tile
  const int kh   = (lane >> 4) << 1;   // K pair select: 0 (lanes 0-15) / 2 (16-31)
  const int row0 = blockIdx.x * BM;
  const int col0 = blockIdx.y * BN;

  v8f acck[4], acch[4];
#pragma unroll
  for (int nt = 0; nt < 4; ++nt) {
#pragma unroll
    for (int i = 0; i < 8; ++i) { acck[nt][i] = 0.0f; acch[nt][i] = 0.0f; }
  }

  for (int k0 = 0; k0 < DXc; k0 += KC) {
    __syncthreads();  // previous chunk's fragments consumed
    // stage x tile: 128 rows x 32 K (8 x b128 per row), coalesced
    for (int i = tid; i < BM * (KC / 4); i += 256) {
      const int r = i >> 3, c4 = (i & 7) << 2;
      copy16_g2l(&x[(size_t)(row0 + r) * DXc + k0 + c4], &As[r * LDA + c4]);
    }
    // stage weight tiles: 32 K x 64 N each (16 x b128 per K row), coalesced
    for (int i = tid; i < KC * (BN / 4); i += 256) {
      const int kk = i >> 4, n4 = (i & 15) << 2;
      const size_t g = (size_t)(k0 + kk) * DHc + col0 + n4;
      copy16_g2l(&Wz[g], &Wzs[kk * BN + n4]);
      copy16_g2l(&Wh[g], &Whs[kk * BN + n4]);
    }
    stage_fence();
    __syncthreads();
#pragma unroll
    for (int kk = 0; kk < KC; kk += 4) {
      // A fragment: 16x4 f32 -> 2 VGPRs (lane = row, half-wave picks K pair)
      const float* ap = &As[(wave * 16 + lh) * LDA + kk + kh];
      v2f a; a.x = ap[0]; a.y = ap[1];
#pragma unroll
      for (int nt = 0; nt < 4; ++nt) {
        const int nn = nt * 16 + lh;
        v2f bzf, bhf;
        bzf.x = Wzs[(kk + kh) * BN + nn];
        bzf.y = Wzs[(kk + kh + 1) * BN + nn];
        bhf.x = Whs[(kk + kh) * BN + nn];
        bhf.y = Whs[(kk + kh + 1) * BN + nn];
        acck[nt] = __builtin_amdgcn_wmma_f32_16x16x4_f32(
            false, a, false, bzf, (short)0, acck[nt], false, false);
        acch[nt] = __builtin_amdgcn_wmma_f32_16x16x4_f32(
            false, a, false, bhf, (short)0, acch[nt], false, false);
      }
    }
  }

  // epilogue: C layout = VGPR i holds M=i (lanes 0-15) / M=8+i (lanes 16-31)
  const int rbase = row0 + wave * 16 + ((lane >> 4) << 3);
#pragma unroll
  for (int nt = 0; nt < 4; ++nt) {
    const int col = col0 + nt * 16 + lh;
    const float bzv = bz[col];
    const float bhv = bh[col];
#pragma unroll
    for (int i = 0; i < 8; ++i) {
      const float kv = acck[nt][i] + bzv;
      const float tv = acch[nt][i] + bhv;
      const size_t idx = (size_t)(rbase + i) * DHc + col;
      coeff[idx] = -softplus_f(kv);                    // log(1-z)
      val[idx]   = -softplus_f(-kv) + log_g_f(tv);     // log z + log g(h~)
    }
  }
}

// ---------- kernel 2: per-chunk segment aggregates ----------
__global__ __launch_bounds__(256) void scan_pass1_kernel(
    const float* __restrict__ coeff, const float* __restrict__ val,
    float* __restrict__ aggA, float* __restrict__ aggV) {
  const int t = blockIdx.x * 256 + threadIdx.x;   // over B*NC*DH
  const int h = t % DHc;
  const int c = (t / DHc) % NC;
  const int b = t / (DHc * NC);
  size_t base = ((size_t)b * SS + (size_t)c * CLEN) * DHc + h;
  float A = 0.0f, V = -INFINITY;
  for (int i = 0; i < CLEN; ++i) {
    const float a = coeff[base];
    const float v = val[base];
    A += a;
    V = logaddexp_f(V + a, v);
    base += DHc;
  }
  aggA[t] = A;
  aggV[t] = V;
}

// ---------- kernel 3: exclusive scan over chunk aggregates ----------
__global__ __launch_bounds__(256) void scan_pass2_kernel(
    const float* __restrict__ h_prev, const float* __restrict__ aggA,
    const float* __restrict__ aggV, float* __restrict__ pref) {
  const int t = blockIdx.x * 256 + threadIdx.x;   // over B*DH
  const int h = t % DHc;
  const int b = t / DHc;
  float running = log_g_f(h_prev[b * DHc + h]);
  for (int c = 0; c < NC; ++c) {
    const size_t idx = ((size_t)b * NC + c) * DHc + h;
    pref[idx] = running;
    running = logaddexp_f(aggA[idx] + running, aggV[idx]);
  }
}

// ---------- kernel 4: replay chunks, write h = exp(log_h) in place ----------
__global__ __launch_bounds__(256) void scan_pass3_kernel(
    const float* __restrict__ coeff, float* __restrict__ val,
    const float* __restrict__ pref) {
  const int t = blockIdx.x * 256 + threadIdx.x;   // over B*NC*DH
  const int h = t % DHc;
  const int c = (t / DHc) % NC;
  const int b = t / (DHc * NC);
  float running = pref[t];
  size_t base = ((size_t)b * SS + (size_t)c * CLEN) * DHc + h;
  for (int i = 0; i < CLEN; ++i) {
    const float a = coeff[base];
    const float v = val[base];
    running = logaddexp_f(a + running, v);
    val[base] = __expf(running);   // read-before-write: safe in place
    base += DHc;
  }
}

// ---------- kernel 5: output GEMM out = h*Wo + bo ----------
__global__ __launch_bounds__(256) void out_gemm_kernel(
    const float* __restrict__ h, const float* __restrict__ Wo,
    const float* __restrict__ bo, float* __restrict__ out) {
  __shared__ float As[BM * LDA];
  __shared__ float Wos[KC * BN];
  const int tid  = threadIdx.x;
  const int wave = tid >> 5;
  const int lane = tid & 31;
  const int lh   = lane & 15;
  const int kh   = (lane >> 4) << 1;
  const int row0 = blockIdx.x * BM;
  const int col0 = blockIdx.y * BN;

  v8f acc[4];
#pragma unroll
  for (int nt = 0; nt < 4; ++nt)
#pragma unroll
    for (int i = 0; i < 8; ++i) acc[nt][i] = 0.0f;

  for (int k0 = 0; k0 < DHc; k0 += KC) {
    __syncthreads();
    for (int i = tid; i < BM * (KC / 4); i += 256) {
      const int r = i >> 3, c4 = (i & 7) << 2;
      copy16_g2l(&h[(size_t)(row0 + r) * DHc + k0 + c4], &As[r * LDA + c4]);
    }
    for (int i = tid; i < KC * (BN / 4); i += 256) {
      const int kk = i >> 4, n4 = (i & 15) << 2;
      copy16_g2l(&Wo[(size_t)(k0 + kk) * DXc + col0 + n4], &Wos[kk * BN + n4]);
    }
    stage_fence();
    __syncthreads();
#pragma unroll
    for (int kk = 0; kk < KC; kk += 4) {
      const float* ap = &As[(wave * 16 + lh) * LDA + kk + kh];
      v2f a; a.x = ap[0]; a.y = ap[1];
#pragma unroll
      for (int nt = 0; nt < 4; ++nt) {
        const int nn = nt * 16 + lh;
        v2f bf;
        bf.x = Wos[(kk + kh) * BN + nn];
        bf.y = Wos[(kk + kh + 1) * BN + nn];
        acc[nt] = __builtin_amdgcn_wmma_f32_16x16x4_f32(
            false, a, false, bf, (short)0, acc[nt], false, false);
      }
    }
  }

  const int rbase = row0 + wave * 16 + ((lane >> 4) << 3);
#pragma unroll
  for (int nt = 0; nt < 4; ++nt) {
    const int col = col0 + nt * 16 + lh;
    const float bov = bo[col];
#pragma unroll
    for (int i = 0; i < 8; ++i) {
      out[(size_t)(rbase + i) * DXc + col] = acc[nt][i] + bov;
    }
  }
}

extern "C" void kernel_launch(void* const* d_in, const int* in_sizes, int n_in,
                              void* d_out, int out_size, void* d_ws, size_t ws_size,
                              hipStream_t stream) {
  const float* x      = (const float*)d_in[0];
  const float* h_prev = (const float*)d_in[1];
  const float* Wz     = (const float*)d_in[2];
  const float* bz     = (const float*)d_in[3];
  const float* Wh     = (const float*)d_in[4];
  const float* bh     = (const float*)d_in[5];
  const float* Wo     = (const float*)d_in[6];
  const float* bo     = (const float*)d_in[7];
  float* out = (float*)d_out;

  // workspace layout (fp32): coeff | val(->h) | aggA | aggV | pref
  const size_t NE = (size_t)BB * SS * DHc;       // 16.7M floats
  const size_t NA = (size_t)BB * NC * DHc;       // 262k floats
  float* coeff = (float*)d_ws;
  float* val   = coeff + NE;
  float* aggA  = val + NE;
  float* aggV  = aggA + NA;
  float* pref  = aggV + NA;

  dim3 gg((BB * SS) / BM, DHc / BN);             // 256 x 8
  gates_gemm_kernel<<<gg, 256, 0, stream>>>(x, Wz, bz, Wh, bh, coeff, val);

  scan_pass1_kernel<<<(BB * NC * DHc) / 256, 256, 0, stream>>>(coeff, val, aggA, aggV);
  scan_pass2_kernel<<<(BB * DHc) / 256, 256, 0, stream>>>(h_prev, aggA, aggV, pref);
  scan_pass3_kernel<<<(BB * NC * DHc) / 256, 256, 0, stream>>>(coeff, val, pref);

  dim3 go((BB * SS) / BM, DXc / BN);             // 256 x 8
  out_gemm_kernel<<<go, 256, 0, stream>>>(val, Wo, bo, out);
}